// GE2ELoss_53317724012670
// MI455X (gfx1250) — compile-verified
//
#include <hip/hip_runtime.h>
#include <math.h>

typedef float v2f __attribute__((ext_vector_type(2)));
typedef float v8f __attribute__((ext_vector_type(8)));

#define N_SPK 1024
#define M_UTT 16
#define D_DIM 512
#define COS_EPS 1e-8f
#define SIM_EPS 1e-6f
#define LOG_EPS 1e-6f
#define HSTRIDE 260   // 256 + 4 pad: 260 mod 64 == 4 -> rows hit distinct bank pairs

// ---------------------------------------------------------------------------
// Kernel 0: zero the scalar output (it is poisoned before timing).
// ---------------------------------------------------------------------------
__global__ void ge2e_zero(float* __restrict__ out) {
    if (threadIdx.x == 0) out[0] = 0.0f;
}

// ---------------------------------------------------------------------------
// Kernel 1: per-speaker preprocessing (unchanged from round 1).
// Block n (256 thr = 16 m-lanes x 16 d-groups of 32):
//   - centroid = mean_m(emb), unit-normalized, stored TRANSPOSED+pair-packed:
//       ws_c[(d>>1)*2048 + n*2 + (d&1)]  (kernel 2 B-frag = one coalesced b64)
//   - diag[n,m]  = dot(e_unit, loo_unit)
//   - inve[n,m]  = 1/max(||e||, 1e-8)
// ---------------------------------------------------------------------------
__global__ __launch_bounds__(256) void ge2e_prep(
    const float* __restrict__ emb,
    float* __restrict__ ws_c,
    float* __restrict__ diag_g,
    float* __restrict__ inve_g) {

    const int n   = blockIdx.x;
    const int tid = threadIdx.x;
    const int m   = tid & 15;
    const int dt  = tid >> 4;

    const float* base = emb + (size_t)n * (M_UTT * D_DIM) + m * D_DIM + dt * 32;

    float ev[32];
#pragma unroll
    for (int j = 0; j < 32; ++j) ev[j] = base[j];

    float c2p = 0.f, e2p = 0.f, l2p = 0.f, elp = 0.f;
    float cent[32];
#pragma unroll
    for (int j = 0; j < 32; ++j) {
        float cs = ev[j];
        cs += __shfl_xor(cs, 1, 32);
        cs += __shfl_xor(cs, 2, 32);
        cs += __shfl_xor(cs, 4, 32);
        cs += __shfl_xor(cs, 8, 32);
        float c = cs * (1.0f / 16.0f);
        cent[j] = c;
        c2p += c * c;
        float loo = (cs - ev[j]) * (1.0f / 15.0f);
        e2p += ev[j] * ev[j];
        l2p += loo * loo;
        elp += ev[j] * loo;
    }

    __shared__ float s_c[16];
    __shared__ float s_cinv;
    __shared__ float s_e2[256], s_l2[256], s_el[256];

    if (m == 0) s_c[dt] = c2p;
    s_e2[tid] = e2p; s_l2[tid] = l2p; s_el[tid] = elp;
    __syncthreads();

    if (tid == 0) {
        float c2 = 0.f;
        for (int k = 0; k < 16; ++k) c2 += s_c[k];
        s_cinv = 1.0f / fmaxf(sqrtf(c2), COS_EPS);
    }
    __syncthreads();

    const float cinv = s_cinv;
    if (m == 0) {
#pragma unroll
        for (int j = 0; j < 32; ++j) {
            int d = dt * 32 + j;
            ws_c[(size_t)(d >> 1) * 2048 + n * 2 + (d & 1)] = cent[j] * cinv;
        }
    }
    if (tid < 16) {
        float e2 = 0.f, l2 = 0.f, el = 0.f;
        for (int k = 0; k < 16; ++k) {
            e2 += s_e2[k * 16 + tid];
            l2 += s_l2[k * 16 + tid];
            el += s_el[k * 16 + tid];
        }
        float einv = 1.0f / fmaxf(sqrtf(e2), COS_EPS);
        float linv = 1.0f / fmaxf(sqrtf(l2), COS_EPS);
        diag_g[n * 16 + tid] = el * einv * linv;
        inve_g[n * 16 + tid] = einv;
    }
}

// ---------------------------------------------------------------------------
// Kernel 2: fused GEMM (v_wmma_f32_16x16x4_f32) + diagonal fix + logsumexp.
// Block b owns TWO speakers (n0=2b, n1=2b+1) -> 32 rows; every B fragment
// feeds 2 WMMAs (halves L2 traffic vs round 1). 16 waves x 64 columns each.
// A tiles staged in LDS in two K-halves of 256 to keep LDS ~35 KB.
// ---------------------------------------------------------------------------
__global__ __launch_bounds__(512) void ge2e_main(
    const float* __restrict__ emb,
    const float* __restrict__ ws_c,
    const float* __restrict__ diag_g,
    const float* __restrict__ inve_g,
    const float* __restrict__ w_p,
    const float* __restrict__ b_p,
    float* __restrict__ out) {

    const int n0     = blockIdx.x * 2;
    const int n1     = n0 + 1;
    const int tid    = threadIdx.x;
    const int lane   = tid & 31;
    const int wave   = tid >> 5;     // 0..15
    const int lanelo = lane & 15;
    const int hi     = lane >> 4;

    __shared__ float s_e[2 * 16 * HSTRIDE];   // [spk][row][col-in-half]
    __shared__ float s_inv[32];               // [spk*16 + m]
    __shared__ float s_diag[32];
    __shared__ float s_part[16][32];

    if (tid < 32) {
        s_inv[tid]  = inve_g[n0 * 16 + tid];
        s_diag[tid] = diag_g[n0 * 16 + tid];
    }

    const float w   = w_p[0];
    const float bsc = b_p[0];
    const int kbase = wave * 64;
    const float2* cp = (const float2*)ws_c;   // pair index p*1024 + k

    v8f acc0[4] = {};
    v8f acc1[4] = {};

    for (int kh = 0; kh < 2; ++kh) {
        __syncthreads();   // also publishes s_inv on first pass
        // Stage 2 speakers x 16 rows x 256 cols of e_unit for this K-half.
        const float* eb = emb + (size_t)n0 * (M_UTT * D_DIM) + kh * 256;
#pragma unroll
        for (int i = 0; i < 16; ++i) {
            int f   = tid + 512 * i;          // 0..8191
            int spk = f >> 12;
            int row = (f >> 8) & 15;
            int col = f & 255;
            s_e[spk * (16 * HSTRIDE) + row * HSTRIDE + col] =
                eb[spk * (M_UTT * D_DIM) + row * D_DIM + col] * s_inv[spk * 16 + row];
        }
        __syncthreads();

        const float* a0p = &s_e[lanelo * HSTRIDE + 2 * hi];
        const float* a1p = a0p + 16 * HSTRIDE;
        for (int d0 = 0; d0 < 256; d0 += 4) {
            float2 av0 = *(const float2*)(a0p + d0);
            float2 av1 = *(const float2*)(a1p + d0);
            v2f a0; a0.x = av0.x; a0.y = av0.y;
            v2f a1; a1.x = av1.x; a1.y = av1.y;
            int p = ((kh * 256 + d0) >> 1) + hi;
#pragma unroll
            for (int ct = 0; ct < 4; ++ct) {
                float2 bv = cp[(size_t)p * 1024 + (kbase + ct * 16 + lanelo)];
                v2f b; b.x = bv.x; b.y = bv.y;
                acc0[ct] = __builtin_amdgcn_wmma_f32_16x16x4_f32(
                    false, a0, false, b, (short)0, acc0[ct], false, false);
                acc1[ct] = __builtin_amdgcn_wmma_f32_16x16x4_f32(
                    false, a1, false, b, (short)0, acc1[ct], false, false);
            }
        }
    }

    // C/D layout: VGPR v -> row v (lanes 0-15) / row v+8 (lanes 16-31), N = lanelo.
    float ex0[8], ex1[8];
#pragma unroll
    for (int v = 0; v < 8; ++v) { ex0[v] = 0.f; ex1[v] = 0.f; }
#pragma unroll
    for (int ct = 0; ct < 4; ++ct) {
        int k = kbase + ct * 16 + lanelo;
#pragma unroll
        for (int v = 0; v < 8; ++v) {
            int row = v + (hi << 3);
            float v0 = acc0[ct][v];
            if (k == n0) v0 = s_diag[row];          // speaker-0 diagonal
            ex0[v] += __expf(fmaf(w, v0 + SIM_EPS, bsc));
            float v1 = acc1[ct][v];
            if (k == n1) v1 = s_diag[16 + row];     // speaker-1 diagonal
            ex1[v] += __expf(fmaf(w, v1 + SIM_EPS, bsc));
        }
    }
#pragma unroll
    for (int v = 0; v < 8; ++v) {
        float s0 = ex0[v], s1 = ex1[v];
        s0 += __shfl_xor(s0, 1, 32);  s1 += __shfl_xor(s1, 1, 32);
        s0 += __shfl_xor(s0, 2, 32);  s1 += __shfl_xor(s1, 2, 32);
        s0 += __shfl_xor(s0, 4, 32);  s1 += __shfl_xor(s1, 4, 32);
        s0 += __shfl_xor(s0, 8, 32);  s1 += __shfl_xor(s1, 8, 32);
        if (lane == 0)  { s_part[wave][v]          = s0; s_part[wave][16 + v]     = s1; }
        if (lane == 16) { s_part[wave][v + 8]      = s0; s_part[wave][16 + v + 8] = s1; }
    }
    __syncthreads();

    if (tid < 32) {
        float tot = 0.f;
#pragma unroll
        for (int wv = 0; wv < 16; ++wv) tot += s_part[wv][tid];
        float neg = __logf(tot + LOG_EPS);
        float pos = fmaf(w, s_diag[tid] + SIM_EPS, bsc);
        float contrib = neg - pos;
        contrib += __shfl_xor(contrib, 1, 32);
        contrib += __shfl_xor(contrib, 2, 32);
        contrib += __shfl_xor(contrib, 4, 32);
        contrib += __shfl_xor(contrib, 8, 32);
        contrib += __shfl_xor(contrib, 16, 32);
        if (tid == 0) atomicAdd(out, contrib);
    }
}

// ---------------------------------------------------------------------------
extern "C" void kernel_launch(void* const* d_in, const int* in_sizes, int n_in,
                              void* d_out, int out_size, void* d_ws, size_t ws_size,
                              hipStream_t stream) {
    const float* emb = (const float*)d_in[0];
    const float* w   = (const float*)d_in[1];
    const float* b   = (const float*)d_in[2];
    float* out = (float*)d_out;

    float* ws   = (float*)d_ws;
    float* ws_c = ws;                      // 1024*512 floats (transposed c_unit)
    float* diag = ws + 524288;             // 16384 floats
    float* inve = ws + 524288 + 16384;     // 16384 floats

    ge2e_zero<<<1, 32, 0, stream>>>(out);
    ge2e_prep<<<N_SPK, 256, 0, stream>>>(emb, ws_c, diag, inve);
    ge2e_main<<<N_SPK / 2, 512, 0, stream>>>(emb, ws_c, diag, inve, w, b, out);
}